// GraphSAGE_14705968022324
// MI455X (gfx1250) — compile-verified
//
#include <hip/hip_runtime.h>
#include <hip/hip_bf16.h>

#define N_NODES 100000
#define N_EDGES 3200000
#define N_GRAPHS 64
#define F_IN 26
#define HID 64
#define N_TILES (N_NODES / 16)   // 6250 exactly

typedef __attribute__((ext_vector_type(2))) float v2f;
typedef __attribute__((ext_vector_type(8))) float v8f;

// Workspace layout (float offsets)
#define OFF_MSG1 ((size_t)0)                          // N_NODES*F_IN
#define OFF_DEG  (OFF_MSG1 + (size_t)N_NODES*F_IN)    // N_NODES
#define OFF_MSG2 (OFF_DEG  + (size_t)N_NODES)         // N_NODES*HID
#define OFF_POOL (OFF_MSG2 + (size_t)N_NODES*HID)     // N_GRAPHS*HID
#define OFF_CNT  (OFF_POOL + (size_t)N_GRAPHS*HID)    // N_GRAPHS
#define OFF_H    (OFF_CNT  + (size_t)N_GRAPHS)        // N_NODES*HID
#define ZERO_FLOATS OFF_H                             // everything before h needs zeroing

__device__ __forceinline__ v8f wmma4(v2f a, v2f b, v8f c) {
    // D = A(16x4 f32) * B(4x16 f32) + C(16x16 f32), exact f32 via matrix core
    return __builtin_amdgcn_wmma_f32_16x16x4_f32(
        /*neg_a=*/false, a, /*neg_b=*/false, b,
        /*c_mod=*/(short)0, c, /*reuse_a=*/false, /*reuse_b=*/false);
}

__global__ __launch_bounds__(256) void zero_ws_kernel(float* __restrict__ p, size_t n) {
    size_t i = (size_t)blockIdx.x * blockDim.x + threadIdx.x;
    size_t stride = (size_t)gridDim.x * blockDim.x;
    for (; i < n; i += stride) p[i] = 0.0f;
}

// Wave per edge: lanes 0..25 carry one input feature each (coalesced gather +
// L2-resident f32 scatter atomics), lane 31 accumulates degree.
__global__ __launch_bounds__(256) void edge_pass1_kernel(
    const float* __restrict__ x, const long long* __restrict__ ei,
    float* __restrict__ msg1, float* __restrict__ deg) {
    const int lane = threadIdx.x & 31;
    long long wid = (long long)blockIdx.x * (blockDim.x >> 5) + (threadIdx.x >> 5);
    long long nw  = (long long)gridDim.x * (blockDim.x >> 5);
    for (long long e = wid; e < N_EDGES; e += nw) {
        const long long s = ei[e];
        const long long d = ei[N_EDGES + e];
        if (lane < F_IN) atomicAdd(&msg1[d * F_IN + lane], x[s * F_IN + lane]);
        if (lane == 31)  atomicAdd(&deg[d], 1.0f);
    }
}

// Wave per edge, 64 hidden features = 2 per lane.
__global__ __launch_bounds__(256) void edge_pass2_kernel(
    const float* __restrict__ h, const long long* __restrict__ ei,
    float* __restrict__ msg2) {
    const int lane = threadIdx.x & 31;
    long long wid = (long long)blockIdx.x * (blockDim.x >> 5) + (threadIdx.x >> 5);
    long long nw  = (long long)gridDim.x * (blockDim.x >> 5);
    for (long long e = wid; e < N_EDGES; e += nw) {
        const long long s = ei[e];
        const long long d = ei[N_EDGES + e];
        atomicAdd(&msg2[d * HID + lane],      h[s * HID + lane]);
        atomicAdd(&msg2[d * HID + 32 + lane], h[s * HID + 32 + lane]);
    }
}

// Layer 1: h = relu(mean1 @ W1l + b1 + x @ W1r). One wave per 16-node tile.
// A layout (16x4 f32): lane l, reg v -> A[l&15][2*(l>>4)+v]
// B layout (4x16 f32): lane l, reg v -> B[2*(l>>4)+v][l&15]
// C/D (16x16 f32):     lane l, reg r -> C[r + 8*(l>>4)][l&15]
__global__ __launch_bounds__(256) void node_layer1_kernel(
    const float* __restrict__ x, const float* __restrict__ msg1,
    const float* __restrict__ deg, const float* __restrict__ W1l,
    const float* __restrict__ b1v, const float* __restrict__ W1r,
    float* __restrict__ h) {
    const int lane = threadIdx.x & 31;
    const int wv   = threadIdx.x >> 5;
    const int tile = blockIdx.x * 8 + wv;     // wave-uniform
    if (tile >= N_TILES) return;              // whole-wave exit: EXEC stays all-1s
    const int n0   = tile * 16;
    const int half = lane >> 4;
    const int col  = lane & 15;
    const int arow = n0 + col;                // node whose row this lane feeds into A
    const float invd = 1.0f / fmaxf(deg[arow], 1.0f);

    v8f acc[4] = { {}, {}, {}, {} };

    // mean1 @ W1l  (K = 26, padded to 28 with zero guards)
    for (int ks = 0; ks < 7; ++ks) {
        const int k0 = ks * 4 + half * 2;
        const int k1 = k0 + 1;
        v2f a;
        a.x = (k0 < F_IN) ? msg1[(size_t)arow * F_IN + k0] * invd : 0.0f;
        a.y = (k1 < F_IN) ? msg1[(size_t)arow * F_IN + k1] * invd : 0.0f;
        for (int t = 0; t < 4; ++t) {
            v2f b;
            b.x = (k0 < F_IN) ? W1l[k0 * HID + t * 16 + col] : 0.0f;
            b.y = (k1 < F_IN) ? W1l[k1 * HID + t * 16 + col] : 0.0f;
            acc[t] = wmma4(a, b, acc[t]);
        }
    }
    // x @ W1r
    for (int ks = 0; ks < 7; ++ks) {
        const int k0 = ks * 4 + half * 2;
        const int k1 = k0 + 1;
        v2f a;
        a.x = (k0 < F_IN) ? x[(size_t)arow * F_IN + k0] : 0.0f;
        a.y = (k1 < F_IN) ? x[(size_t)arow * F_IN + k1] : 0.0f;
        for (int t = 0; t < 4; ++t) {
            v2f b;
            b.x = (k0 < F_IN) ? W1r[k0 * HID + t * 16 + col] : 0.0f;
            b.y = (k1 < F_IN) ? W1r[k1 * HID + t * 16 + col] : 0.0f;
            acc[t] = wmma4(a, b, acc[t]);
        }
    }
    // bias + relu + store
    for (int t = 0; t < 4; ++t) {
        const float bias = b1v[t * 16 + col];
        for (int r = 0; r < 8; ++r) {
            const int m = r + half * 8;
            const float v = fmaxf(acc[t][r] + bias, 0.0f);
            h[(size_t)(n0 + m) * HID + t * 16 + col] = v;
        }
    }
}

// Layer 2 fused with global mean pool: h2 = mean2@W2l + b2 + h@W2r,
// accumulated straight into pooled[g][:] via f32 atomics (tiny, L2-resident).
__global__ __launch_bounds__(256) void node_layer2_pool_kernel(
    const float* __restrict__ msg2, const float* __restrict__ deg,
    const float* __restrict__ h, const float* __restrict__ W2l,
    const float* __restrict__ b2v, const float* __restrict__ W2r,
    const long long* __restrict__ batch, float* __restrict__ pooled,
    float* __restrict__ counts) {
    const int lane = threadIdx.x & 31;
    const int wv   = threadIdx.x >> 5;
    const int tile = blockIdx.x * 8 + wv;     // wave-uniform
    if (tile >= N_TILES) return;
    const int n0   = tile * 16;
    const int half = lane >> 4;
    const int col  = lane & 15;
    const int arow = n0 + col;
    const float invd = 1.0f / fmaxf(deg[arow], 1.0f);

    // one count per node (partial-EXEC vector op, happens before any WMMA issue)
    if (lane < 16) atomicAdd(&counts[(int)batch[n0 + lane]], 1.0f);

    v8f acc[4] = { {}, {}, {}, {} };

    // mean2 @ W2l  (K = 64, exact)
    for (int ks = 0; ks < 16; ++ks) {
        const int k0 = ks * 4 + half * 2;
        const int k1 = k0 + 1;
        v2f a;
        a.x = msg2[(size_t)arow * HID + k0] * invd;
        a.y = msg2[(size_t)arow * HID + k1] * invd;
        for (int t = 0; t < 4; ++t) {
            v2f b;
            b.x = W2l[k0 * HID + t * 16 + col];
            b.y = W2l[k1 * HID + t * 16 + col];
            acc[t] = wmma4(a, b, acc[t]);
        }
    }
    // h @ W2r
    for (int ks = 0; ks < 16; ++ks) {
        const int k0 = ks * 4 + half * 2;
        const int k1 = k0 + 1;
        v2f a;
        a.x = h[(size_t)arow * HID + k0];
        a.y = h[(size_t)arow * HID + k1];
        for (int t = 0; t < 4; ++t) {
            v2f b;
            b.x = W2r[k0 * HID + t * 16 + col];
            b.y = W2r[k1 * HID + t * 16 + col];
            acc[t] = wmma4(a, b, acc[t]);
        }
    }
    // bias + scatter into per-graph pooled sums
    for (int t = 0; t < 4; ++t) {
        const float bias = b2v[t * 16 + col];
        for (int r = 0; r < 8; ++r) {
            const int m = r + half * 8;
            const int g = (int)batch[n0 + m];
            atomicAdd(&pooled[g * HID + t * 16 + col], acc[t][r] + bias);
        }
    }
}

// out[g][j] = blin[j] + sum_k (pooled[g][k]/max(cnt,1)) * Wlin[k][j]
__global__ __launch_bounds__(256) void head_kernel(
    const float* __restrict__ pooled, const float* __restrict__ counts,
    const float* __restrict__ Wlin, const float* __restrict__ blin,
    float* __restrict__ out) {
    const int idx = blockIdx.x * blockDim.x + threadIdx.x;
    if (idx >= N_GRAPHS * F_IN) return;
    const int g = idx / F_IN;
    const int j = idx % F_IN;
    const float inv = 1.0f / fmaxf(counts[g], 1.0f);
    float s = blin[j];
    for (int k = 0; k < HID; ++k)
        s += pooled[g * HID + k] * inv * Wlin[k * F_IN + j];
    out[idx] = s;
}

extern "C" void kernel_launch(void* const* d_in, const int* in_sizes, int n_in,
                              void* d_out, int out_size, void* d_ws, size_t ws_size,
                              hipStream_t stream) {
    const float*     x     = (const float*)d_in[0];
    const long long* ei    = (const long long*)d_in[1];
    const long long* batch = (const long long*)d_in[2];
    const float*     W1l   = (const float*)d_in[3];
    const float*     b1v   = (const float*)d_in[4];
    const float*     W1r   = (const float*)d_in[5];
    const float*     W2l   = (const float*)d_in[6];
    const float*     b2v   = (const float*)d_in[7];
    const float*     W2r   = (const float*)d_in[8];
    const float*     Wlin  = (const float*)d_in[9];
    const float*     blin  = (const float*)d_in[10];
    float* out = (float*)d_out;

    float* ws     = (float*)d_ws;
    float* msg1   = ws + OFF_MSG1;
    float* deg    = ws + OFF_DEG;
    float* msg2   = ws + OFF_MSG2;
    float* pooled = ws + OFF_POOL;
    float* counts = ws + OFF_CNT;
    float* h      = ws + OFF_H;

    zero_ws_kernel<<<2048, 256, 0, stream>>>(ws, ZERO_FLOATS);
    edge_pass1_kernel<<<20000, 256, 0, stream>>>(x, ei, msg1, deg);
    node_layer1_kernel<<<(N_TILES + 7) / 8, 256, 0, stream>>>(
        x, msg1, deg, W1l, b1v, W1r, h);
    edge_pass2_kernel<<<20000, 256, 0, stream>>>(h, ei, msg2);
    node_layer2_pool_kernel<<<(N_TILES + 7) / 8, 256, 0, stream>>>(
        msg2, deg, h, W2l, b2v, W2r, batch, pooled, counts);
    head_kernel<<<(N_GRAPHS * F_IN + 255) / 256, 256, 0, stream>>>(
        pooled, counts, Wlin, blin, out);
}